// MLPAttention_80358838108542
// MI455X (gfx1250) — compile-verified
//
#include <hip/hip_runtime.h>

// ---------------------------------------------------------------------------
// MLP-additive-attention, algebraically collapsed for MI455X (gfx1250):
//   softmax_k(sq[b,q] + sk[b,k]) == softmax_k(sk[b,k])   (shift invariance)
// so out[b,q,:] = sum_k softmax(mask(sk[b,:]))[k] * value[b,k,:], broadcast
// over q. query/Wq cancel entirely. Bandwidth floor ~192MB => ~8us @23.3TB/s.
// ctx GEMV runs on V_WMMA_F32_16X16X4_F32; weight row staged to LDS via the
// CDNA5 async global->LDS DMA (ASYNCcnt) path.
// ---------------------------------------------------------------------------

typedef __attribute__((ext_vector_type(2))) float v2f;
typedef __attribute__((ext_vector_type(8))) float v8f;
typedef __attribute__((address_space(3))) float lds_float;

#define NEG_VAL (-1.0e6f)
#define BN 16
#define QN 1024
#define KN 1024
#define DN 1024
#define UN 1024

// ---- Kernel 1: wk_v[d] = sum_u Wk[d,u] * v_w[u]  (block per row) ----------
__global__ void k_wkv(const float* __restrict__ Wk, const float* __restrict__ v_w,
                      float* __restrict__ wk_v) {
  __shared__ float red[256];
  const int row = blockIdx.x;
  const int t = threadIdx.x;
  const float* wrow = Wk + (size_t)row * UN;
  float acc = 0.f;
#pragma unroll
  for (int j = 0; j < UN / 256; ++j) {
    const int u = t + j * 256;
    acc += wrow[u] * v_w[u];
  }
  red[t] = acc;
  __syncthreads();
  for (int s = 128; s > 0; s >>= 1) {
    if (t < s) red[t] += red[t + s];
    __syncthreads();
  }
  if (t == 0) wk_v[row] = red[0];
}

// ---- Kernel 2: sk[b,k] = key_in[b,k,:] . wk_v  (one wave32 per row) -------
__global__ void k_sk(const float* __restrict__ key_in, const float* __restrict__ wk_v,
                     float* __restrict__ sk) {
  const int lane = threadIdx.x & 31;
  const int wave = threadIdx.x >> 5;
  const int row = blockIdx.x * 8 + wave;          // 0..B*K-1
  const float* krow = key_in + (size_t)row * DN;
  // CDNA5 global_prefetch_b8 path (speculative; OOB silently dropped)
  __builtin_prefetch(krow + DN, 0, 1);
  float acc = 0.f;
#pragma unroll
  for (int j = 0; j < DN / 128; ++j) {
    const int d = j * 128 + lane * 4;
    const float4 kv = *(const float4*)(krow + d);
    const float4 wv = *(const float4*)(wk_v + d);
    acc += kv.x * wv.x + kv.y * wv.y + kv.z * wv.z + kv.w * wv.w;
  }
#pragma unroll
  for (int off = 16; off >= 1; off >>= 1)
    acc += __shfl_xor(acc, off, 32);
  if (lane == 0) sk[row] = acc;
}

// ---- Kernel 3: per-batch masked softmax over sk -> w ----------------------
__global__ void k_softmax(const float* __restrict__ sk, const int* __restrict__ valid_len,
                          float* __restrict__ w) {
  __shared__ float red[256];
  const int b = blockIdx.x;
  const int t = threadIdx.x;
  const int vl = valid_len[b];
  float loc[KN / 256];
  float mx = NEG_VAL;
#pragma unroll
  for (int j = 0; j < KN / 256; ++j) {
    const int k = t + j * 256;
    const float s = (k < vl) ? sk[b * KN + k] : NEG_VAL;
    loc[j] = s;
    mx = fmaxf(mx, s);
  }
  red[t] = mx;
  __syncthreads();
  for (int s = 128; s > 0; s >>= 1) {
    if (t < s) red[t] = fmaxf(red[t], red[t + s]);
    __syncthreads();
  }
  mx = red[0];
  __syncthreads();
  float psum = 0.f;
#pragma unroll
  for (int j = 0; j < KN / 256; ++j) {
    loc[j] = __expf(loc[j] - mx);
    psum += loc[j];
  }
  red[t] = psum;
  __syncthreads();
  for (int s = 128; s > 0; s >>= 1) {
    if (t < s) red[t] += red[t + s];
    __syncthreads();
  }
  const float inv = 1.0f / red[0];
#pragma unroll
  for (int j = 0; j < KN / 256; ++j)
    w[b * KN + t + j * 256] = loc[j] * inv;
}

// ---- Kernel 4: ctx[b,:] = w[b,:] @ value[b,:,:]  via V_WMMA_F32_16X16X4_F32
// grid = BN*8 blocks; each block: one batch slice, 8 waves x one 16-wide
// d-tile each => 1024 concurrent waves streaming `value` (64MB read).
// Weight row (4KB) staged to LDS with GLOBAL_LOAD_ASYNC_TO_LDS_B128
// (ASYNCcnt-tracked CDNA5 DMA). The LDS destination pointer is passed to the
// asm as an addrspace(3) operand (== LDS byte offset VGPR) so the shared
// array ESCAPES: the "memory" clobber then forces the compiler to keep the
// allocation and re-load wlds after the barrier (ds ops feeding WMMA A).
// A tile = weight row replicated over the 16 M-rows (all C rows equal; row 0
// is read out). A K-layout per ISA 16x4 table: v0 = K0/K2 lane halves,
// v1 = K1/K3; B striped symmetrically.
__global__ void k_ctx(const float* __restrict__ w, const float* __restrict__ value,
                      float* __restrict__ ctx) {
  __shared__ float wlds[KN];
  const int b = blockIdx.x >> 3;        // batch
  const int g = blockIdx.x & 7;         // d-tile group within batch
  const int t = threadIdx.x;            // 256 threads = 8 waves

  // Async DMA: each lane moves 16B global -> LDS; 256 lanes == 4KB row.
  {
    lds_float* ldst = (lds_float*)&wlds[t * 4];          // LDS byte offset VGPR
    const float* gsrc = w + (size_t)b * KN + t * 4;
    asm volatile("global_load_async_to_lds_b128 %0, %1, off"
                 :: "v"(ldst), "v"(gsrc) : "memory");
    asm volatile("s_wait_asynccnt 0x0" ::: "memory");
  }
  __syncthreads();

  const int lane = t & 31;
  const int wave = t >> 5;              // 0..7
  const int half = lane >> 4;           // 0: K0/K1, 1: K2/K3
  const int col  = lane & 15;           // N within the 16-wide d tile
  const int tile = g * 8 + wave;        // 0..63
  const int d0 = tile * 16;
  const float* vbase = value + (size_t)b * KN * DN;

  v8f c = {0.f, 0.f, 0.f, 0.f, 0.f, 0.f, 0.f, 0.f};
  for (int kb = 0; kb < KN; kb += 4) {
    const int k0 = kb + half * 2;
    v2f a;  a.x = wlds[k0];  a.y = wlds[k0 + 1];          // ds_load_2addr_b64
    const float* vp = vbase + (size_t)k0 * DN + d0 + col;
    v2f bm; bm.x = vp[0];    bm.y = vp[DN];               // rows k0, k0+1
    c = __builtin_amdgcn_wmma_f32_16x16x4_f32(
        /*neg_a=*/false, a, /*neg_b=*/false, bm,
        /*c_mod=*/(short)0, c, /*reuse_a=*/false, /*reuse_b=*/false);
  }
  // C layout: VGPR0, lanes 0-15 hold M=0, N=lane.
  if (lane < 16) ctx[b * DN + d0 + lane] = c[0];
}

// ---- Kernel 5: out[b,q,:] = ctx[b,:]  (float4 broadcast stores) -----------
__global__ void k_bcast(const float* __restrict__ ctx, float* __restrict__ out) {
  const size_t idx4 = (size_t)blockIdx.x * blockDim.x + threadIdx.x; // 4M float4s
  const int d4 = (int)(idx4 & (DN / 4 - 1));        // 0..255
  const size_t bq = idx4 >> 8;
  const int b = (int)(bq >> 10);                    // /Q
  const float4 v = *(const float4*)(ctx + ((size_t)b << 10) + (d4 << 2));
  *(float4*)(out + (idx4 << 2)) = v;
}

// ---------------------------------------------------------------------------
extern "C" void kernel_launch(void* const* d_in, const int* in_sizes, int n_in,
                              void* d_out, int out_size, void* d_ws, size_t ws_size,
                              hipStream_t stream) {
  (void)in_sizes; (void)n_in; (void)out_size; (void)ws_size;
  // setup_inputs order: 0=query (unused!), 1=key_in, 2=value, 3=Wq (unused!),
  //                     4=Wk, 5=v_w, 6=valid_len
  const float* key_in    = (const float*)d_in[1];
  const float* value     = (const float*)d_in[2];
  const float* Wk        = (const float*)d_in[4];
  const float* v_w       = (const float*)d_in[5];
  const int*   valid_len = (const int*)d_in[6];
  float* out = (float*)d_out;

  float* ws   = (float*)d_ws;
  float* wk_v = ws;                       // 1024
  float* sk   = wk_v + DN;                // 16384
  float* wgt  = sk + BN * KN;             // 16384
  float* ctx  = wgt + BN * KN;            // 16384   (~200 KB total)

  k_wkv    <<<DN,              256, 0, stream>>>(Wk, v_w, wk_v);
  k_sk     <<<BN * KN / 8,     256, 0, stream>>>(key_in, wk_v, sk);
  k_softmax<<<BN,              256, 0, stream>>>(sk, valid_len, wgt);
  k_ctx    <<<BN * 8,          256, 0, stream>>>(wgt, value, ctx);
  k_bcast  <<<BN * QN * (DN/4) / 256, 256, 0, stream>>>(ctx, out);
}